// RF_CRATE_25013889531995
// MI455X (gfx1250) — compile-verified
//
#include <hip/hip_runtime.h>
#include <math.h>

#define DIMN 512
#define NSEQ 512
#define BATCH 8
#define HEADS 8
#define DHEAD 64
#define DEPTH 6
#define PLANE (BATCH * NSEQ * DIMN)   // 2,097,152 floats per real/imag plane
#define EPSLN 1e-5f
#define ATT_SCALE 0.125f              // DH^-0.5 = 1/8

typedef __attribute__((ext_vector_type(2))) float v2f;
typedef __attribute__((ext_vector_type(8))) float v8f;

static __device__ __forceinline__ v8f wmma4(v2f a, v2f b, v8f c) {
  // D = A(16x4 f32) x B(4x16 f32) + C(16x16 f32); NEG bits must be 0 for f32.
  return __builtin_amdgcn_wmma_f32_16x16x4_f32(false, a, false, b, (short)0, c,
                                               false, false);
}

static __device__ __forceinline__ v2f ld2(const float* p) {
  float2 t = *(const float2*)p;
  v2f r;
  r.x = t.x;
  r.y = t.y;
  return r;
}

static __device__ __forceinline__ v8f splat8(float v) {
  v8f r;
#pragma unroll
  for (int i = 0; i < 8; ++i) r[i] = v;
  return r;
}

// ---------------------------------------------------------------------------
// LayerNorm over one real plane: out = (v-mean)*rsqrt(var+eps)*g + b
// One 256-thread block per row of 512.
// ---------------------------------------------------------------------------
__global__ __launch_bounds__(256) void cln_kernel(const float* __restrict__ v,
                                                  const float* __restrict__ g,
                                                  const float* __restrict__ bta,
                                                  float* __restrict__ out) {
  __shared__ float s1[256];
  __shared__ float s2[256];
  const int row = blockIdx.x;
  const int tid = threadIdx.x;
  const float x0 = v[row * DIMN + tid];
  const float x1 = v[row * DIMN + 256 + tid];
  s1[tid] = x0 + x1;
  s2[tid] = x0 * x0 + x1 * x1;
  __syncthreads();
#pragma unroll
  for (int off = 128; off > 0; off >>= 1) {
    if (tid < off) {
      s1[tid] += s1[tid + off];
      s2[tid] += s2[tid + off];
    }
    __syncthreads();
  }
  const float mean = s1[0] * (1.0f / DIMN);
  const float var = s2[0] * (1.0f / DIMN) - mean * mean;
  const float rstd = rsqrtf(var + EPSLN);
  out[row * DIMN + tid] = (x0 - mean) * rstd * g[tid] + bta[tid];
  out[row * DIMN + 256 + tid] = (x1 - mean) * rstd * g[tid + 256] + bta[tid + 256];
}

// ---------------------------------------------------------------------------
// Complex GEMM: C[m,n] = sum_k A[m,k] * W[n,k]   (A: 4096x512, W: 512x512)
// One wave32 per 32x32 complex tile: 2x2 subtiles of 16x16, so each K=4 step
// issues 8 fragment loads and 16 f32 WMMAs (fragments reused across subtiles).
// mode 0: store C       (qkv projection)
// mode 1: x += C + bias (out-projection + residual)
// mode 2: ISTA + CReLU  (ff block; A is also the z term)
// ---------------------------------------------------------------------------
__global__ __launch_bounds__(32) void cgemm_kernel(
    const float* __restrict__ Ar, const float* __restrict__ Ai,
    const float* __restrict__ Wr, const float* __restrict__ Wi,
    float* __restrict__ Cr, float* __restrict__ Ci, int mode,
    const float* __restrict__ biasR, const float* __restrict__ biasI,
    float* __restrict__ Xr, float* __restrict__ Xi,
    const float* __restrict__ stepP, const float* __restrict__ lamP) {
  const int lane = threadIdx.x;
  const int half = lane >> 4;
  const int lrow = lane & 15;
  const int hoff = half * 2;  // A/B frag: lanes 16-31 hold K+2,K+3
  const int m0 = blockIdx.y * 32;
  const int n0 = blockIdx.x * 32;

  const float* arp0 = Ar + (m0 + lrow) * DIMN + hoff;
  const float* arp1 = arp0 + 16 * DIMN;
  const float* aip0 = Ai + (m0 + lrow) * DIMN + hoff;
  const float* aip1 = aip0 + 16 * DIMN;
  const float* wrp0 = Wr + (n0 + lrow) * DIMN + hoff;
  const float* wrp1 = wrp0 + 16 * DIMN;
  const float* wip0 = Wi + (n0 + lrow) * DIMN + hoff;
  const float* wip1 = wip0 + 16 * DIMN;

  v8f cr[2][2], ci[2][2];
#pragma unroll
  for (int u = 0; u < 2; ++u)
#pragma unroll
    for (int t = 0; t < 2; ++t) {
      cr[u][t] = splat8(0.f);
      ci[u][t] = splat8(0.f);
    }

#pragma unroll 1
  for (int k = 0; k < DIMN; k += 4) {
    if ((k & 63) == 0 && k + 64 < DIMN) {
      // gfx1250 global_prefetch_b8: pull next K-panel lines toward the WGP.
      __builtin_prefetch(arp0 + k + 64, 0, 3);
      __builtin_prefetch(aip0 + k + 64, 0, 3);
      __builtin_prefetch(wrp0 + k + 64, 0, 3);
      __builtin_prefetch(wip0 + k + 64, 0, 3);
    }
    v2f a_r[2], a_i[2], b_r[2], b_i[2], nb_i[2];
    a_r[0] = ld2(arp0 + k);
    a_r[1] = ld2(arp1 + k);
    a_i[0] = ld2(aip0 + k);
    a_i[1] = ld2(aip1 + k);
    b_r[0] = ld2(wrp0 + k);
    b_r[1] = ld2(wrp1 + k);
    b_i[0] = ld2(wip0 + k);
    b_i[1] = ld2(wip1 + k);
    nb_i[0] = -b_i[0];
    nb_i[1] = -b_i[1];
#pragma unroll
    for (int u = 0; u < 2; ++u) {
#pragma unroll
      for (int t = 0; t < 2; ++t) {
        cr[u][t] = wmma4(a_r[u], b_r[t], cr[u][t]);   // += Ar*Wr
        cr[u][t] = wmma4(a_i[u], nb_i[t], cr[u][t]);  // -= Ai*Wi
        ci[u][t] = wmma4(a_r[u], b_i[t], ci[u][t]);   // += Ar*Wi
        ci[u][t] = wmma4(a_i[u], b_r[t], ci[u][t]);   // += Ai*Wr
      }
    }
  }

  float ps = 0.f, pl = 0.f;
  if (mode == 2) {
    ps = logf(1.0f + expf(stepP[0]));  // softplus(step)
    pl = logf(1.0f + expf(lamP[0]));   // softplus(lambda)
  }

#pragma unroll
  for (int u = 0; u < 2; ++u) {
#pragma unroll
    for (int t = 0; t < 2; ++t) {
#pragma unroll
      for (int r = 0; r < 8; ++r) {
        const int row = u * 16 + r + half * 8;  // D layout: lanes 16-31 -> M=r+8
        const int col = t * 16 + lrow;
        const int g = (m0 + row) * DIMN + n0 + col;
        const float vr = cr[u][t][r];
        const float vi = ci[u][t][r];
        if (mode == 0) {
          Cr[g] = vr;
          Ci[g] = vi;
        } else if (mode == 1) {
          Xr[g] = Xr[g] + vr + biasR[n0 + col];
          Xi[g] = Xi[g] + vi + biasI[n0 + col];
        } else {
          const float ur = Ar[g] + ps * vr - ps * pl;
          const float ui = Ai[g] + ps * vi;
          Xr[g] = fmaxf(ur, 0.f);
          Xi[g] = fmaxf(ui, 0.f);
        }
      }
    }
  }
}

// ---------------------------------------------------------------------------
// Complex magnitude-softmax attention (flash-style, online softmax).
// One wave32 per (b, h, 16-query tile). q = k = v = W head slice.
// dots = q * conj(k) * scale; attn = softmax(|dots|) * dots/|dots|.
// ---------------------------------------------------------------------------
__global__ __launch_bounds__(32) void cattn_kernel(const float* __restrict__ Wr,
                                                   const float* __restrict__ Wi,
                                                   float* __restrict__ AOr,
                                                   float* __restrict__ AOi) {
  __shared__ float sPr[256];
  __shared__ float sPi[256];

  const int lane = threadIdx.x;
  const int half = lane >> 4;
  const int lrow = lane & 15;
  const int hoff = half * 2;
  const int q0 = blockIdx.x * 16;
  const int bh = blockIdx.y;
  const int b = bh >> 3;
  const int h = bh & 7;
  const int base = b * NSEQ * DIMN + h * DHEAD;  // element (n,d) at base+n*512+d

  // Q fragments (16 rows x 64 d), kept in registers across the whole m-loop.
  v2f qr[16], qi[16];
#pragma unroll
  for (int s = 0; s < 16; ++s) {
    const int idx = base + (q0 + lrow) * DIMN + s * 4 + hoff;
    qr[s] = ld2(Wr + idx);
    qi[s] = ld2(Wi + idx);
  }

  v8f mrun = splat8(-INFINITY);
  v8f lrun = splat8(0.f);
  v8f oR[4], oI[4];
#pragma unroll
  for (int nt = 0; nt < 4; ++nt) {
    oR[nt] = splat8(0.f);
    oI[nt] = splat8(0.f);
  }

#pragma unroll 1
  for (int mt = 0; mt < NSEQ / 16; ++mt) {
    const int mm0 = mt * 16;
    // dots tile: re = qr*kr + qi*ki ; im = qi*kr - qr*ki   (conj on k)
    v8f dR = splat8(0.f);
    v8f dI = splat8(0.f);
#pragma unroll
    for (int s = 0; s < 16; ++s) {
      const int kidx = base + (mm0 + lrow) * DIMN + s * 4 + hoff;
      v2f br = ld2(Wr + kidx);
      v2f bi = ld2(Wi + kidx);
      v2f nbi = -bi;
      dR = wmma4(qr[s], br, dR);
      dR = wmma4(qi[s], bi, dR);
      dI = wmma4(qi[s], br, dI);
      dI = wmma4(qr[s], nbi, dI);
    }

    // Online softmax over |dots| per query row (16 cols live in one 16-lane
    // half of the wave: xor masks 1,2,4,8 never cross the half boundary).
#pragma unroll
    for (int r = 0; r < 8; ++r) {
      const float re = dR[r] * ATT_SCALE;
      const float im = dI[r] * ATT_SCALE;
      const float mag = sqrtf(re * re + im * im);
      float rmax = mag;
      rmax = fmaxf(rmax, __shfl_xor(rmax, 1));
      rmax = fmaxf(rmax, __shfl_xor(rmax, 2));
      rmax = fmaxf(rmax, __shfl_xor(rmax, 4));
      rmax = fmaxf(rmax, __shfl_xor(rmax, 8));
      const float mnew = fmaxf(mrun[r], rmax);
      const float corr = expf(mrun[r] - mnew);
      const float p = expf(mag - mnew);
      float pr, pi;
      if (mag > 0.f) {
        const float inv = 1.f / mag;
        pr = p * re * inv;   // p * exp(i*angle)
        pi = p * im * inv;
      } else {
        pr = p;              // angle(0) == 0 -> factor 1
        pi = 0.f;
      }
      float rsum = p;
      rsum += __shfl_xor(rsum, 1);
      rsum += __shfl_xor(rsum, 2);
      rsum += __shfl_xor(rsum, 4);
      rsum += __shfl_xor(rsum, 8);
      lrun[r] = lrun[r] * corr + rsum;
      mrun[r] = mnew;
#pragma unroll
      for (int nt = 0; nt < 4; ++nt) {
        oR[nt][r] *= corr;
        oI[nt][r] *= corr;
      }
      const int prow = r + half * 8;  // D-layout row for this lane
      sPr[prow * 16 + lrow] = pr;
      sPi[prow * 16 + lrow] = pi;
    }
    __syncthreads();

    // out += P(16x16 complex) x V(16x64 complex); P read back as A-fragments.
#pragma unroll
    for (int s2 = 0; s2 < 4; ++s2) {
      const int koff = s2 * 4 + hoff;
      v2f par, pai;
      par.x = sPr[lrow * 16 + koff];
      par.y = sPr[lrow * 16 + koff + 1];
      pai.x = sPi[lrow * 16 + koff];
      pai.y = sPi[lrow * 16 + koff + 1];
      v2f npai = -pai;
#pragma unroll
      for (int nt = 0; nt < 4; ++nt) {
        const int r0 = base + (mm0 + koff) * DIMN + nt * 16 + lrow;
        v2f br, bi;
        br.x = Wr[r0];
        br.y = Wr[r0 + DIMN];
        bi.x = Wi[r0];
        bi.y = Wi[r0 + DIMN];
        oR[nt] = wmma4(par, br, oR[nt]);
        oR[nt] = wmma4(npai, bi, oR[nt]);  // -= Pi*Vi
        oI[nt] = wmma4(par, bi, oI[nt]);
        oI[nt] = wmma4(pai, br, oI[nt]);
      }
    }
    __syncthreads();
  }

#pragma unroll
  for (int r = 0; r < 8; ++r) {
    const float inv = 1.f / lrun[r];
    const int row = r + half * 8;
#pragma unroll
    for (int nt = 0; nt < 4; ++nt) {
      const int g = base + (q0 + row) * DIMN + nt * 16 + lrow;
      AOr[g] = oR[nt][r] * inv;
      AOi[g] = oI[nt][r] * inv;
    }
  }
}

// ---------------------------------------------------------------------------
__global__ __launch_bounds__(256) void init_copy(const float* __restrict__ xr_in,
                                                 const float* __restrict__ xi_in,
                                                 float* __restrict__ xr,
                                                 float* __restrict__ xi) {
  const int i = blockIdx.x * 256 + threadIdx.x;
  if (i < PLANE) {
    xr[i] = xr_in[i];
    xi[i] = xi_in[i];
  }
}

// ---------------------------------------------------------------------------
extern "C" void kernel_launch(void* const* d_in, const int* in_sizes, int n_in,
                              void* d_out, int out_size, void* d_ws,
                              size_t ws_size, hipStream_t stream) {
  (void)in_sizes; (void)n_in; (void)out_size; (void)ws_size;
  // setup_inputs() dict order:
  const float* x_r    = (const float*)d_in[0];
  const float* x_i    = (const float*)d_in[1];
  const float* ln1_gr = (const float*)d_in[2];
  const float* ln1_gi = (const float*)d_in[3];
  const float* ln2_gr = (const float*)d_in[4];
  const float* ln2_gi = (const float*)d_in[5];
  const float* ln1_br = (const float*)d_in[6];
  const float* ln1_bi = (const float*)d_in[7];
  const float* ln2_br = (const float*)d_in[8];
  const float* ln2_bi = (const float*)d_in[9];
  const float* qkv_wr = (const float*)d_in[10];
  const float* qkv_wi = (const float*)d_in[11];
  const float* out_wr = (const float*)d_in[12];
  const float* out_wi = (const float*)d_in[13];
  const float* out_br = (const float*)d_in[14];
  const float* out_bi = (const float*)d_in[15];
  const float* ff_wr  = (const float*)d_in[16];
  const float* ff_wi  = (const float*)d_in[17];
  const float* stepP  = (const float*)d_in[18];
  const float* lamP   = (const float*)d_in[19];

  // x lives directly in d_out as [2, B, N, DIM] (real plane, imag plane).
  float* xr = (float*)d_out;
  float* xi = xr + PLANE;
  float* ws = (float*)d_ws;
  float* bAr = ws;              // z1 / attention-out
  float* bAi = ws + PLANE;
  float* bBr = ws + 2 * PLANE;  // qkv-out w / z2
  float* bBi = ws + 3 * PLANE;

  init_copy<<<PLANE / 256, 256, 0, stream>>>(x_r, x_i, xr, xi);

  const dim3 gGemm(DIMN / 32, (BATCH * NSEQ) / 32);  // 16 x 128 tiles of 32x32
  const dim3 gAttn(NSEQ / 16, BATCH * HEADS);        // 32 x 64

  for (int l = 0; l < DEPTH; ++l) {
    const int wOff = l * DIMN * DIMN;
    const int pOff = l * DIMN;

    // z1 = ComplexLayerNorm(x)
    cln_kernel<<<BATCH * NSEQ, 256, 0, stream>>>(xr, ln1_gr + pOff, ln1_br + pOff, bAr);
    cln_kernel<<<BATCH * NSEQ, 256, 0, stream>>>(xi, ln1_gi + pOff, ln1_bi + pOff, bAi);

    // w = z1 @ Wqkv^T  (complex)
    cgemm_kernel<<<gGemm, 32, 0, stream>>>(bAr, bAi, qkv_wr + wOff, qkv_wi + wOff,
                                           bBr, bBi, 0, nullptr, nullptr,
                                           nullptr, nullptr, nullptr, nullptr);

    // ao = polar-softmax attention over w (per b,h)
    cattn_kernel<<<gAttn, 32, 0, stream>>>(bBr, bBi, bAr, bAi);

    // x = ao @ Wout^T + bout + x
    cgemm_kernel<<<gGemm, 32, 0, stream>>>(bAr, bAi, out_wr + wOff, out_wi + wOff,
                                           nullptr, nullptr, 1, out_br + pOff,
                                           out_bi + pOff, xr, xi, nullptr, nullptr);

    // z2 = ComplexLayerNorm(x)
    cln_kernel<<<BATCH * NSEQ, 256, 0, stream>>>(xr, ln2_gr + pOff, ln2_br + pOff, bBr);
    cln_kernel<<<BATCH * NSEQ, 256, 0, stream>>>(xi, ln2_gi + pOff, ln2_bi + pOff, bBi);

    // x = CReLU(z2 + ps*(z2 @ Wff^T) - ps*pl)
    cgemm_kernel<<<gGemm, 32, 0, stream>>>(bBr, bBi, ff_wr + wOff, ff_wi + wOff,
                                           nullptr, nullptr, 2, nullptr, nullptr,
                                           xr, xi, stepP + l, lamP + l);
  }
}